// GCN_43044162240574
// MI455X (gfx1250) — compile-verified
//
#include <hip/hip_runtime.h>

typedef __attribute__((ext_vector_type(2))) float v2f;
typedef __attribute__((ext_vector_type(8))) float v8f;

#define N_NODES   50000
#define N_EDGES   800000
#define D_IN      128
#define HIDDEN    64
#define N_CLASSES 2

// ---------------------------------------------------------------------------
// Init: deg=1.0 (self-loop), acc1=0 (vectorized), out=0
// ---------------------------------------------------------------------------
__global__ void gcn_init(float* __restrict__ deg, float4* __restrict__ acc1v,
                         float2* __restrict__ outv) {
    int i = blockIdx.x * blockDim.x + threadIdx.x;
    if (i < N_NODES * (HIDDEN / 4)) acc1v[i] = make_float4(0.f, 0.f, 0.f, 0.f);
    if (i < N_NODES) {
        deg[i]  = 1.0f;                       // self-loop contributes 1
        outv[i] = make_float2(0.f, 0.f);      // N_CLASSES == 2
    }
}

// ---------------------------------------------------------------------------
// Degree accumulation over edges (dst side), then dinv = rsqrt(deg)
// ---------------------------------------------------------------------------
__global__ void gcn_deg(const int* __restrict__ ei, float* __restrict__ deg) {
    int e = blockIdx.x * blockDim.x + threadIdx.x;
    if (e >= N_EDGES) return;
    atomicAdd(&deg[ei[N_EDGES + e]], 1.0f);
}

__global__ void gcn_dinv(float* __restrict__ deg) {
    int i = blockIdx.x * blockDim.x + threadIdx.x;
    if (i >= N_NODES) return;
    float d = deg[i];
    deg[i] = (d > 0.0f) ? rsqrtf(d) : 0.0f;
}

// ---------------------------------------------------------------------------
// GEMM1: h1[50000,64] = x[50000,128] @ W1[128,64]   (fp32 WMMA 16x16x4)
// One wave32 per 16x16 output tile. 3125 M-tiles x 4 N-tiles, all exact.
//
// fp32 A 16x4 layout (ISA 7.12.2): lanes 0-15 -> M=0..15, VGPR0 holds K=0
// (K=2 for upper half-lanes), VGPR1 holds K=1 (K=3 upper). B (4x16) mirrors
// with N striped across lanes; C/D: VGPR j = row j (lower half) / j+8 (upper).
// ---------------------------------------------------------------------------
__global__ void gcn_gemm1(const float* __restrict__ x,
                          const float* __restrict__ W1,
                          float* __restrict__ h1) {
    const int lane = threadIdx.x & 31;
    const int wave = threadIdx.x >> 5;
    const int tile = blockIdx.x * 4 + wave;          // 12500 tiles total
    const int mt = tile >> 2;                         // 0..3124
    const int nt = tile & 3;                          // 0..3
    const int r    = lane & 15;
    const int half = lane >> 4;

    const float* __restrict__ Arow = x + (mt * 16 + r) * D_IN;
    const int bcol = nt * 16 + r;

    v8f c = {};
#pragma unroll
    for (int k = 0; k < D_IN; k += 4) {
        const int kk = k + 2 * half;
        v2f a = *(const v2f*)(Arow + kk);             // A[M=r][kk], A[M=r][kk+1]
        v2f b;
        b.x = W1[kk * HIDDEN + bcol];                 // B[kk][col]
        b.y = W1[(kk + 1) * HIDDEN + bcol];           // B[kk+1][col]
        c = __builtin_amdgcn_wmma_f32_16x16x4_f32(
                false, a, false, b, (short)0, c, false, false);
    }

#pragma unroll
    for (int j = 0; j < 8; ++j)
        h1[(mt * 16 + j + 8 * half) * HIDDEN + bcol] = c[j];
}

// ---------------------------------------------------------------------------
// Edge aggregation layer 1: acc1[dst,f] += h1[src,f] * dinv[src]*dinv[dst]
// One thread per (edge, 4-feature group): 16 threads/edge, b128 gather of h1.
// h1 (12.8MB) and acc1 (12.8MB) are L2-resident on the 192MB L2.
// ---------------------------------------------------------------------------
__global__ void gcn_agg1(const int* __restrict__ ei,
                         const float* __restrict__ dinv,
                         const float* __restrict__ h1,
                         float* __restrict__ acc1) {
    unsigned t = blockIdx.x * blockDim.x + threadIdx.x;
    if (t >= (unsigned)N_EDGES * (HIDDEN / 4)) return;
    int e  = t >> 4;                // / 16
    int f4 = (t & 15) << 2;         // feature group base
    int s = ei[e];
    int d = ei[N_EDGES + e];
    float norm = dinv[s] * dinv[d];
    float4 hv = *(const float4*)(h1 + s * HIDDEN + f4);
    float* dst = acc1 + d * HIDDEN + f4;
    atomicAdd(dst + 0, hv.x * norm);
    atomicAdd(dst + 1, hv.y * norm);
    atomicAdd(dst + 2, hv.z * norm);
    atomicAdd(dst + 3, hv.w * norm);
}

// ---------------------------------------------------------------------------
// Epilogue 1: fold self-loop (h1[i]*dinv^2), add bias, ReLU (in place -> a1)
// Vectorized x4.
// ---------------------------------------------------------------------------
__global__ void gcn_relu1(const float* __restrict__ dinv,
                          const float4* __restrict__ h1v,
                          const float4* __restrict__ b1v,
                          float4* __restrict__ acc1v) {
    int i = blockIdx.x * blockDim.x + threadIdx.x;
    if (i >= N_NODES * (HIDDEN / 4)) return;
    int node = i >> 4;
    int f4   = i & 15;
    float di = dinv[node];
    float d2 = di * di;
    float4 a = acc1v[i];
    float4 h = h1v[i];
    float4 b = b1v[f4];
    float vx = a.x + h.x * d2 + b.x;
    float vy = a.y + h.y * d2 + b.y;
    float vz = a.z + h.z * d2 + b.z;
    float vw = a.w + h.w * d2 + b.w;
    acc1v[i] = make_float4(vx > 0.f ? vx : 0.f, vy > 0.f ? vy : 0.f,
                           vz > 0.f ? vz : 0.f, vw > 0.f ? vw : 0.f);
}

// ---------------------------------------------------------------------------
// GEMM2: h2[50000,2] = a1[50000,64] @ W2[64,2]  (fp32 WMMA, N padded to 16)
// One wave per 16-row tile; B columns >= N_CLASSES are zero-filled.
// ---------------------------------------------------------------------------
__global__ void gcn_gemm2(const float* __restrict__ a1,
                          const float* __restrict__ W2,
                          float* __restrict__ h2) {
    const int lane = threadIdx.x & 31;
    const int wave = threadIdx.x >> 5;
    const int mt = blockIdx.x * 4 + wave;
    if (mt >= N_NODES / 16) return;                   // 3125 tiles
    const int r    = lane & 15;
    const int half = lane >> 4;

    const float* __restrict__ Arow = a1 + (mt * 16 + r) * HIDDEN;
    const bool valid_col = (r < N_CLASSES);

    v8f c = {};
#pragma unroll
    for (int k = 0; k < HIDDEN; k += 4) {
        const int kk = k + 2 * half;
        v2f a = *(const v2f*)(Arow + kk);
        v2f b;
        b.x = valid_col ? W2[kk * N_CLASSES + r] : 0.0f;
        b.y = valid_col ? W2[(kk + 1) * N_CLASSES + r] : 0.0f;
        c = __builtin_amdgcn_wmma_f32_16x16x4_f32(
                false, a, false, b, (short)0, c, false, false);
    }

    if (valid_col) {
#pragma unroll
        for (int j = 0; j < 8; ++j)
            h2[(mt * 16 + j + 8 * half) * N_CLASSES + r] = c[j];
    }
}

// ---------------------------------------------------------------------------
// Edge aggregation layer 2: one thread per edge, float2 gather, 2 atomics.
// ---------------------------------------------------------------------------
__global__ void gcn_agg2(const int* __restrict__ ei,
                         const float* __restrict__ dinv,
                         const float* __restrict__ h2,
                         float* __restrict__ out) {
    int e = blockIdx.x * blockDim.x + threadIdx.x;
    if (e >= N_EDGES) return;
    int s = ei[e];
    int d = ei[N_EDGES + e];
    float norm = dinv[s] * dinv[d];
    float2 hv = *(const float2*)(h2 + s * N_CLASSES);
    atomicAdd(out + d * N_CLASSES + 0, hv.x * norm);
    atomicAdd(out + d * N_CLASSES + 1, hv.y * norm);
}

// ---------------------------------------------------------------------------
// Epilogue 2: out += self-loop + bias   (one thread per node, float2)
// ---------------------------------------------------------------------------
__global__ void gcn_final(const float* __restrict__ dinv,
                          const float2* __restrict__ h2v,
                          const float* __restrict__ b2,
                          float2* __restrict__ outv) {
    int i = blockIdx.x * blockDim.x + threadIdx.x;
    if (i >= N_NODES) return;
    float di = dinv[i];
    float d2 = di * di;
    float2 h = h2v[i];
    float2 o = outv[i];
    outv[i] = make_float2(o.x + h.x * d2 + b2[0],
                          o.y + h.y * d2 + b2[1]);
}

// ---------------------------------------------------------------------------
extern "C" void kernel_launch(void* const* d_in, const int* in_sizes, int n_in,
                              void* d_out, int out_size, void* d_ws, size_t ws_size,
                              hipStream_t stream) {
    const float* x  = (const float*)d_in[0];
    const int*   ei = (const int*)d_in[1];     // [2, E] int32
    const float* W1 = (const float*)d_in[2];
    const float* b1 = (const float*)d_in[3];
    const float* W2 = (const float*)d_in[4];
    const float* b2 = (const float*)d_in[5];
    float* out = (float*)d_out;

    // Workspace layout (floats): deg/dinv[N] | h1[N*64] | acc1/a1[N*64] | h2[N*2]
    // Offsets: 0, 200000B, 13.0MB, 25.8MB -- all 16B-aligned.
    float* deg  = (float*)d_ws;
    float* h1   = deg  + N_NODES;
    float* acc1 = h1   + (size_t)N_NODES * HIDDEN;
    float* h2   = acc1 + (size_t)N_NODES * HIDDEN;

    const int T = 256;

    // 1. init (grid sized for the widest touch: N*16 float4 stores)
    gcn_init<<<(N_NODES * (HIDDEN / 4) + T - 1) / T, T, 0, stream>>>(
        deg, (float4*)acc1, (float2*)out);
    // 2. degrees
    gcn_deg<<<(N_EDGES + T - 1) / T, T, 0, stream>>>(ei, deg);
    // 3. dinv
    gcn_dinv<<<(N_NODES + T - 1) / T, T, 0, stream>>>(deg);
    // 4. GEMM1 (WMMA): 12500 tiles, 4 waves per 128-thread block
    gcn_gemm1<<<N_NODES / 16, 128, 0, stream>>>(x, W1, h1);
    // 5. scatter layer 1: 16 threads/edge, float4 path
    gcn_agg1<<<((unsigned)N_EDGES * (HIDDEN / 4) + T - 1) / T, T, 0, stream>>>(
        ei, deg, h1, acc1);
    // 6. bias + self-loop + relu (x4 vectorized)
    gcn_relu1<<<(N_NODES * (HIDDEN / 4) + T - 1) / T, T, 0, stream>>>(
        deg, (const float4*)h1, (const float4*)b1, (float4*)acc1);
    // 7. GEMM2 (WMMA): 3125 tiles, 4 waves/block
    gcn_gemm2<<<(N_NODES / 16 + 3) / 4, 128, 0, stream>>>(acc1, W2, h2);
    // 8. scatter layer 2: 1 thread/edge
    gcn_agg2<<<(N_EDGES + T - 1) / T, T, 0, stream>>>(ei, deg, h2, out);
    // 9. self-loop + bias epilogue
    gcn_final<<<(N_NODES + T - 1) / T, T, 0, stream>>>(
        deg, (const float2*)h2, b2, (float2*)out);
}